// AnomalyScorer_72189810311354
// MI455X (gfx1250) — compile-verified
//
#include <hip/hip_runtime.h>
#include <hip/hip_bf16.h>

typedef __attribute__((ext_vector_type(2))) float v2f;
typedef __attribute__((ext_vector_type(8))) float v8f;

#define HID   256
#define BETA  1.0f
#define MU    0.5f

// One 256-thread block = 8 waves; each wave scores 16 edges via
// V_WMMA_F32_16X16X4_F32 accumulating D = C * C^T over the 256-feature dim,
// then reads diag(D) = ||a*h_u + b*h_v||^2 per edge.
__global__ __launch_bounds__(256) void AnomalyScorer_wmma_kernel(
    const float* __restrict__ h,
    const int*   __restrict__ us,
    const int*   __restrict__ vs,
    const float* __restrict__ wsw,
    const float* __restrict__ av,
    const float* __restrict__ bv,
    float* __restrict__ out,
    int n_edges)
{
    __shared__ __align__(16) float s_a[HID];
    __shared__ __align__(16) float s_b[HID];
    for (int i = threadIdx.x; i < HID; i += 256) {
        s_a[i] = av[i];
        s_b[i] = bv[i];
    }
    __syncthreads();

    const int lane = threadIdx.x & 31;
    const int wave = threadIdx.x >> 5;
    const int m    = lane & 15;   // edge slot within this wave's 16 edges
    const int half = lane >> 4;   // 0 -> K%4 in {0,1}; 1 -> K%4 in {2,3}

    const int edge0 = (blockIdx.x * 8 + wave) * 16;
    const int e     = edge0 + m;
    const int ec    = (e < n_edges) ? e : (n_edges - 1);  // clamp: keep EXEC full

    const size_t ru = (size_t)us[ec] * HID;
    const size_t rv = (size_t)vs[ec] * HID;
    const float* __restrict__ hu = h + ru;
    const float* __restrict__ hv = h + rv;

    // f32 WMMA A-operand layout (16x4, K chunks of 4):
    //   VGPR0: lanes 0-15 -> K=0, lanes 16-31 -> K=2
    //   VGPR1: lanes 0-15 -> K=1, lanes 16-31 -> K=3
    // For B = A^T the register image is identical -> feed the same v2f twice.
    v8f acc0 = {};
    v8f acc1 = {};
    const int fbase = 2 * half;

    #pragma unroll 8
    for (int c = 0; c < HID / 8; ++c) {
        {
            const int f = 8 * c + fbase;           // even K-chunk
            v2f xu = *(const v2f*)(hu + f);
            v2f xv = *(const v2f*)(hv + f);
            v2f aa = *(const v2f*)(s_a + f);
            v2f bb = *(const v2f*)(s_b + f);
            v2f cc;
            cc.x = aa.x * xu.x + bb.x * xv.x;
            cc.y = aa.y * xu.y + bb.y * xv.y;
            acc0 = __builtin_amdgcn_wmma_f32_16x16x4_f32(
                false, cc, false, cc, (short)0, acc0, false, false);
        }
        {
            const int f = 8 * c + 4 + fbase;       // odd K-chunk
            v2f xu = *(const v2f*)(hu + f);
            v2f xv = *(const v2f*)(hv + f);
            v2f aa = *(const v2f*)(s_a + f);
            v2f bb = *(const v2f*)(s_b + f);
            v2f cc;
            cc.x = aa.x * xu.x + bb.x * xv.x;
            cc.y = aa.y * xu.y + bb.y * xv.y;
            acc1 = __builtin_amdgcn_wmma_f32_16x16x4_f32(
                false, cc, false, cc, (short)0, acc1, false, false);
        }
    }

    v8f acc;
    #pragma unroll
    for (int r = 0; r < 8; ++r) acc[r] = acc0[r] + acc1[r];

    // Diagonal extraction.
    // C/D layout: VGPR r, lanes 0-15 -> D[M=r][N=lane]; lanes 16-31 -> D[M=r+8][N=lane-16].
    // D[m][m] for m=0..7  lives in lane m,    component m.
    // D[m][m] for m=8..15 lives in lane m+16, component m-8.
    const int sel = lane & 7;
    float dsel = acc[0];
    #pragma unroll
    for (int r = 1; r < 8; ++r) dsel = (sel == r) ? acc[r] : dsel;

    const int src  = (m < 8) ? m : (m + 16);
    const float norm_sq = __shfl(dsel, src, 32);

    if (half == 0 && e < n_edges) {
        const float x   = BETA * (norm_sq - MU);
        const float sig = 1.0f / (1.0f + __expf(-x));
        out[e] = wsw[e] * sig;
    }
}

extern "C" void kernel_launch(void* const* d_in, const int* in_sizes, int n_in,
                              void* d_out, int out_size, void* d_ws, size_t ws_size,
                              hipStream_t stream) {
    const float* h  = (const float*)d_in[0];
    const int*   us = (const int*)  d_in[1];
    const int*   vs = (const int*)  d_in[2];
    const float* ws = (const float*)d_in[3];
    const float* a  = (const float*)d_in[4];
    const float* b  = (const float*)d_in[5];
    float* out = (float*)d_out;

    const int n_edges = in_sizes[1];
    const int edges_per_block = 8 * 16;  // 8 waves x 16 edges
    const int blocks = (n_edges + edges_per_block - 1) / edges_per_block;

    AnomalyScorer_wmma_kernel<<<blocks, 256, 0, stream>>>(
        h, us, vs, ws, a, b, out, n_edges);
}